// Net_7206955123270
// MI455X (gfx1250) — compile-verified
//
#include <hip/hip_runtime.h>
#include <math.h>

#define F_IN 128
#define F_OUT 16

typedef __attribute__((ext_vector_type(2))) float v2f;
typedef __attribute__((ext_vector_type(8))) float v8f;

// ---------------------------------------------------------------------------
// deg[i] = 1 (self loop)
__global__ void k_init_deg(float* __restrict__ deg, int n) {
    int i = blockIdx.x * blockDim.x + threadIdx.x;
    if (i < n) deg[i] = 1.0f;
}

// deg[dst[e]] += 1 for every edge
__global__ void k_deg_accum(const int* __restrict__ dst, float* __restrict__ deg, int e) {
    int i = blockIdx.x * blockDim.x + threadIdx.x;
    if (i < e) atomicAdd(&deg[dst[i]], 1.0f);
}

// dis[i] = rsqrt(deg[i])   (deg >= 1 always)
__global__ void k_dis(const float* __restrict__ deg, float* __restrict__ dis, int n) {
    int i = blockIdx.x * blockDim.x + threadIdx.x;
    if (i < n) dis[i] = rsqrtf(deg[i]);
}

// hout[i][f] = dis[i]^2 * hin[i][f]   (self-loop term, also initializes hout)
__global__ void k_selfloop(const float* __restrict__ hin, const float* __restrict__ dis,
                           float* __restrict__ hout, int n) {
    int i = blockIdx.x * blockDim.x + threadIdx.x;
    if (i < n * F_IN) {
        int node = i >> 7;               // /128
        float d = dis[node];
        hout[i] = d * d * hin[i];
    }
}

// One wave per edge: hout[dst] += dis[src]*dis[dst] * hin[src]
// Lane l handles features [4l, 4l+4) : one coalesced float4 load + 4 f32 atomics.
__global__ void k_edge_scatter(const int* __restrict__ src, const int* __restrict__ dst,
                               const float* __restrict__ dis,
                               const float* __restrict__ hin, float* __restrict__ hout,
                               int e) {
    int gid  = blockIdx.x * blockDim.x + threadIdx.x;
    int edge = gid >> 5;
    int lane = gid & 31;
    if (edge >= e) return;
    int s = src[edge];
    int d = dst[edge];
    float norm = dis[s] * dis[d];
    const float4* hs = (const float4*)(hin + (size_t)s * F_IN);
    float4 v = hs[lane];                 // features 4l..4l+3, fully coalesced
    float* o = hout + (size_t)d * F_IN + lane * 4;
    atomicAdd(o + 0, norm * v.x);
    atomicAdd(o + 1, norm * v.y);
    atomicAdd(o + 2, norm * v.z);
    atomicAdd(o + 3, norm * v.w);
}

// ---------------------------------------------------------------------------
// One wave per 16-row tile: logits = h[16x128] @ W[128x16] + b via 32 steps of
// V_WMMA_F32_16X16X4_F32, then fused log-softmax across the 16 logits (which
// live across 16 lanes of each accumulator VGPR) using shfl_xor butterflies.
//
// Fragment layout (ISA 7.12.2, wave32):
//   A 16x4 f32 : lane holds row M=lane&15; VGPR0=K(+0/+2 by half), VGPR1=K+1(/+3)
//   B 4x16 f32 : VGPR0 holds K=(0|2 by half), N=lane&15; VGPR1 = K+1
//   C/D 16x16  : VGPR r -> row (r | r+8 by half), N=lane&15
__global__ void k_gemm_logsoftmax(const float* __restrict__ h,
                                  const float* __restrict__ W,
                                  const float* __restrict__ bias,
                                  float* __restrict__ out,
                                  int ntiles, int n) {
    int wave = threadIdx.x >> 5;
    int tile = blockIdx.x * (blockDim.x >> 5) + wave;
    if (tile >= ntiles) return;          // wave-uniform: EXEC all-1 inside

    int lane = threadIdx.x & 31;
    int half = lane >> 4;                // 0: lanes 0-15, 1: lanes 16-31
    int l16  = lane & 15;
    int koff = half * 2;

    int arow = tile * 16 + l16;          // A-fragment source row (M = lane&15)
    if (arow >= n) arow = n - 1;         // clamp (n is a multiple of 16 here)
    const float* ap = h + (size_t)arow * F_IN;

    v8f c = {0.f, 0.f, 0.f, 0.f, 0.f, 0.f, 0.f, 0.f};
#pragma unroll
    for (int k0 = 0; k0 < F_IN; k0 += 4) {
        v2f a, b;
        a.x = ap[k0 + koff];
        a.y = ap[k0 + koff + 1];
        b.x = W[(k0 + koff) * F_OUT + l16];
        b.y = W[(k0 + koff + 1) * F_OUT + l16];
        c = __builtin_amdgcn_wmma_f32_16x16x4_f32(
                /*neg_a=*/false, a, /*neg_b=*/false, b,
                /*c_mod=*/(short)0, c, /*reuse_a=*/false, /*reuse_b=*/false);
    }

    float bv = bias[l16];                // bias depends on N = lane&15 only

#pragma unroll
    for (int r = 0; r < 8; ++r) {
        float v = c[r] + bv;
        // max over the 16 lanes holding this row (masks 1..8 stay in-half)
        float m = v;
        m = fmaxf(m, __shfl_xor(m, 1, 32));
        m = fmaxf(m, __shfl_xor(m, 2, 32));
        m = fmaxf(m, __shfl_xor(m, 4, 32));
        m = fmaxf(m, __shfl_xor(m, 8, 32));
        float ex = __expf(v - m);
        float s = ex;
        s += __shfl_xor(s, 1, 32);
        s += __shfl_xor(s, 2, 32);
        s += __shfl_xor(s, 4, 32);
        s += __shfl_xor(s, 8, 32);
        float res = v - m - __logf(s);
        int orow = tile * 16 + half * 8 + r;   // D row mapping
        if (orow < n) out[(size_t)orow * F_OUT + l16] = res;
    }
}

// ---------------------------------------------------------------------------
extern "C" void kernel_launch(void* const* d_in, const int* in_sizes, int n_in,
                              void* d_out, int out_size, void* d_ws, size_t ws_size,
                              hipStream_t stream) {
    const float* x  = (const float*)d_in[0];   // [N, 128]
    const int*   ei = (const int*)d_in[1];     // [2, E]
    const float* W  = (const float*)d_in[2];   // [128, 16]
    const float* b  = (const float*)d_in[3];   // [16]
    float* out = (float*)d_out;                // [N, 16]

    const int n = in_sizes[0] / F_IN;
    const int e = in_sizes[1] / 2;
    const int* src = ei;
    const int* dst = ei + e;

    float* ws  = (float*)d_ws;
    float* deg = ws;
    float* dis = deg + n;
    float* h1  = dis + n;
    float* h2  = h1 + (size_t)n * F_IN;

    const int T = 256;
    dim3 blk(T);

    // degree + normalization
    k_init_deg <<<(n + T - 1) / T, blk, 0, stream>>>(deg, n);
    k_deg_accum<<<(e + T - 1) / T, blk, 0, stream>>>(dst, deg, e);
    k_dis      <<<(n + T - 1) / T, blk, 0, stream>>>(deg, dis, n);

    const int nf_blocks = (n * F_IN + T - 1) / T;
    const long long ethreads = (long long)e * 32;
    const int es_blocks = (int)((ethreads + T - 1) / T);

    // hop 1: x -> h1
    k_selfloop    <<<nf_blocks, blk, 0, stream>>>(x, dis, h1, n);
    k_edge_scatter<<<es_blocks, blk, 0, stream>>>(src, dst, dis, x, h1, e);
    // hop 2: h1 -> h2
    k_selfloop    <<<nf_blocks, blk, 0, stream>>>(h1, dis, h2, n);
    k_edge_scatter<<<es_blocks, blk, 0, stream>>>(src, dst, dis, h1, h2, e);

    // dense linear + log_softmax (WMMA f32 16x16x4, one wave per 16-row tile)
    const int ntiles = (n + 15) / 16;
    const int waves_per_block = T / 32;
    k_gemm_logsoftmax<<<(ntiles + waves_per_block - 1) / waves_per_block, blk, 0, stream>>>(
        h2, W, b, out, ntiles, n);
}